// Patch_54631984005224
// MI455X (gfx1250) — compile-verified
//
#include <hip/hip_runtime.h>

#define EPS_F 1e-5f
constexpr int Bn = 128, Sn = 512, Dn = 512;

typedef __attribute__((ext_vector_type(2))) float v2f;
typedef __attribute__((ext_vector_type(8))) float v8f;

// ---------------------------------------------------------------------------
// Kernel 1: ori[b,d,s] = (x[b,s,d] - rm[f]) * g[f]*rsqrt(rv[f]+eps) + beta[f]
//           with f = d*S + s.  LDS-tiled transpose for coalescing both sides.
// ---------------------------------------------------------------------------
__global__ __launch_bounds__(256) void bn_transpose_kernel(
    const float* __restrict__ x, const float* __restrict__ g,
    const float* __restrict__ bta, const float* __restrict__ rm,
    const float* __restrict__ rv, float* __restrict__ ori)
{
    __shared__ float tile[32][33];
    const int b  = blockIdx.z;
    const int s0 = blockIdx.x * 32, d0 = blockIdx.y * 32;
    const int tx = threadIdx.x, ty = threadIdx.y;          // block (32, 8)

    const float* xb = x + (size_t)b * Sn * Dn;
#pragma unroll
    for (int i = 0; i < 4; ++i) {
        int sl = ty + 8 * i;
        tile[tx][sl] = xb[(size_t)(s0 + sl) * Dn + (d0 + tx)];  // coalesced in d
    }
    __syncthreads();

    float* orib = ori + (size_t)b * Dn * Sn;
#pragma unroll
    for (int i = 0; i < 4; ++i) {
        int dl = ty + 8 * i;
        int f  = (d0 + dl) * Sn + (s0 + tx);
        float v = tile[dl][tx];
        v = (v - rm[f]) * (g[f] * rsqrtf(rv[f] + EPS_F)) + bta[f];
        orib[(size_t)(d0 + dl) * Sn + (s0 + tx)] = v;           // coalesced in s
    }
}

// ---------------------------------------------------------------------------
// Kernel 2: sequential patch recurrence, one thread per (b,d) row.
// ---------------------------------------------------------------------------
template <int P>
__global__ __launch_bounds__(256) void scan_kernel(
    const float* __restrict__ ori,
    const float* __restrict__ g1, const float* __restrict__ b1,
    const float* __restrict__ rm1, const float* __restrict__ rv1,
    const float* __restrict__ Wa, const float* __restrict__ ba,
    float* __restrict__ outb)
{
    const int r = blockIdx.x * blockDim.x + threadIdx.x;    // [0, B*D)
    const int d = r & (Dn - 1);
    const float* row  = ori  + (size_t)r * Sn;
    float*       orow = outb + (size_t)r * Sn;

    float sc[P], sh[P], bias[P], wa[P][P];
#pragma unroll
    for (int k = 0; k < P; ++k) {
        int f = d * P + k;
        float s = g1[f] * rsqrtf(rv1[f] + EPS_F);
        sc[k]   = s;
        sh[k]   = b1[f] - rm1[f] * s;
        bias[k] = ba[k];
    }
#pragma unroll
    for (int q = 0; q < P; ++q)
#pragma unroll
        for (int k = 0; k < P; ++k) wa[q][k] = Wa[q * P + k];

    float prev[P];
#pragma unroll
    for (int k = 0; k < P; ++k) { prev[k] = row[k]; orow[k] = prev[k]; }

    const int NP = Sn / P;
    for (int i = 1; i < NP; ++i) {
        float nv[P], h[P];
#pragma unroll
        for (int k = 0; k < P; ++k) nv[k] = prev[k] * sc[k] + sh[k];
#pragma unroll
        for (int q = 0; q < P; ++q) {
            float a = bias[q];
#pragma unroll
            for (int k = 0; k < P; ++k) a = fmaf(nv[k], wa[q][k], a);
            h[q] = fmaxf(a, 0.f);
        }
#pragma unroll
        for (int k = 0; k < P; ++k) {
            float t = h[k] + row[i * P + k];
            orow[i * P + k] = t;
            prev[k] = t;
        }
    }
}

// ---------------------------------------------------------------------------
// Kernel 3: y[b,q,m] = sum_s A[b,m,s] * Wl[q,s] + bl[q]  via f32 WMMA 16x16x4.
// Block = 256 threads (8 waves) -> 128(M) x 64(N) tile; each wave computes a
// 32x32 tile (2x2 register blocking -> 4 WMMA per 4 ds_load_b64, 1:1 ratio).
// K staged in chunks of 32, pair-interleaved so each fragment = 1 ds_load_b64:
//   lane L (half=L/16, lm=L%16): A[m=lm][k=k4+2*half+{0,1}] (ISA 16x4 f32),
//                                B[k=k4+2*half+{0,1}][n=lm].
// ---------------------------------------------------------------------------
__global__ __launch_bounds__(256) void wmma_gemm_kernel(
    const float* __restrict__ A,    // [B, D, S]
    const float* __restrict__ Wl,   // [Q, S]
    const float* __restrict__ bl,   // [Q]
    float* __restrict__ Y,          // [B, Q, D]
    int Q)
{
    __shared__ float2 lds_a2[16 * 128];  // [k2][m] : pair (A[m][2k2], A[m][2k2+1]) 16KB
    __shared__ float2 lds_b2[16 * 64];   // [k2][n] : pair (B[2k2][n], B[2k2+1][n])  8KB

    const int b   = blockIdx.z;
    const int n0  = blockIdx.x * 64;
    const int m0  = blockIdx.y * 128;
    const int tid = threadIdx.x;
    const int lane = tid & 31, wave = tid >> 5;
    const int mt = (wave & 3) * 32;      // wave's M sub-tile (32 rows)
    const int nt = (wave >> 2) * 32;     // wave's N sub-tile (32 cols)
    const int half = lane >> 4, lm = lane & 15;

    const float* Ab = A + (size_t)b * Dn * Sn;

    v8f acc00, acc01, acc10, acc11;
    const float bias0 = bl[n0 + nt + lm];
    const float bias1 = bl[n0 + nt + 16 + lm];
#pragma unroll
    for (int i = 0; i < 8; ++i) {
        acc00[i] = bias0; acc10[i] = bias0;
        acc01[i] = bias1; acc11[i] = bias1;
    }

    // cooperative staging assignments
    const int a_m  = tid >> 1;           // 0..127
    const int a_k0 = (tid & 1) * 16;     // 0, 16
    const int b_n  = tid >> 2;           // 0..63
    const int b_k0 = (tid & 3) * 8;      // 0, 8, 16, 24

    const float* Arow = Ab + (size_t)(m0 + a_m) * Sn;
    const float* Brow = Wl + (size_t)(n0 + b_n) * Sn;

    for (int kb = 0; kb < Sn; kb += 32) {
        float4 av0 = *(const float4*)(Arow + kb + a_k0);
        float4 av1 = *(const float4*)(Arow + kb + a_k0 + 4);
        float4 av2 = *(const float4*)(Arow + kb + a_k0 + 8);
        float4 av3 = *(const float4*)(Arow + kb + a_k0 + 12);
        float4 bv0 = *(const float4*)(Brow + kb + b_k0);
        float4 bv1 = *(const float4*)(Brow + kb + b_k0 + 4);

        __syncthreads();   // protect LDS from overwrite while waves still compute
        lds_a2[((a_k0 >> 1) + 0) * 128 + a_m] = make_float2(av0.x, av0.y);
        lds_a2[((a_k0 >> 1) + 1) * 128 + a_m] = make_float2(av0.z, av0.w);
        lds_a2[((a_k0 >> 1) + 2) * 128 + a_m] = make_float2(av1.x, av1.y);
        lds_a2[((a_k0 >> 1) + 3) * 128 + a_m] = make_float2(av1.z, av1.w);
        lds_a2[((a_k0 >> 1) + 4) * 128 + a_m] = make_float2(av2.x, av2.y);
        lds_a2[((a_k0 >> 1) + 5) * 128 + a_m] = make_float2(av2.z, av2.w);
        lds_a2[((a_k0 >> 1) + 6) * 128 + a_m] = make_float2(av3.x, av3.y);
        lds_a2[((a_k0 >> 1) + 7) * 128 + a_m] = make_float2(av3.z, av3.w);
        lds_b2[((b_k0 >> 1) + 0) * 64 + b_n]  = make_float2(bv0.x, bv0.y);
        lds_b2[((b_k0 >> 1) + 1) * 64 + b_n]  = make_float2(bv0.z, bv0.w);
        lds_b2[((b_k0 >> 1) + 2) * 64 + b_n]  = make_float2(bv1.x, bv1.y);
        lds_b2[((b_k0 >> 1) + 3) * 64 + b_n]  = make_float2(bv1.z, bv1.w);
        __syncthreads();

#pragma unroll
        for (int k4 = 0; k4 < 32; k4 += 4) {
            int k2 = (k4 >> 1) + half;
            float2 af0 = lds_a2[k2 * 128 + mt + lm];
            float2 af1 = lds_a2[k2 * 128 + mt + 16 + lm];
            float2 bf0 = lds_b2[k2 * 64 + nt + lm];
            float2 bf1 = lds_b2[k2 * 64 + nt + 16 + lm];
            v2f a0; a0[0] = af0.x; a0[1] = af0.y;
            v2f a1; a1[0] = af1.x; a1[1] = af1.y;
            v2f b0; b0[0] = bf0.x; b0[1] = bf0.y;
            v2f b1; b1[0] = bf1.x; b1[1] = bf1.y;
            acc00 = __builtin_amdgcn_wmma_f32_16x16x4_f32(
                false, a0, false, b0, (short)0, acc00, false, false);
            acc01 = __builtin_amdgcn_wmma_f32_16x16x4_f32(
                false, a0, false, b1, (short)0, acc01, false, false);
            acc10 = __builtin_amdgcn_wmma_f32_16x16x4_f32(
                false, a1, false, b0, (short)0, acc10, false, false);
            acc11 = __builtin_amdgcn_wmma_f32_16x16x4_f32(
                false, a1, false, b1, (short)0, acc11, false, false);
        }
    }

    // store: within a tile, lane holds (m = 8*half + r, n = lm), r = 0..7
    const int q0 = n0 + nt + lm;
    const int q1 = q0 + 16;
    float* ybase = Y + (size_t)b * Q * Dn;

    float* yp = ybase + (size_t)q0 * Dn + (m0 + mt + 8 * half);
    *(float4*)(yp)     = make_float4(acc00[0], acc00[1], acc00[2], acc00[3]);
    *(float4*)(yp + 4) = make_float4(acc00[4], acc00[5], acc00[6], acc00[7]);

    yp = ybase + (size_t)q1 * Dn + (m0 + mt + 8 * half);
    *(float4*)(yp)     = make_float4(acc01[0], acc01[1], acc01[2], acc01[3]);
    *(float4*)(yp + 4) = make_float4(acc01[4], acc01[5], acc01[6], acc01[7]);

    yp = ybase + (size_t)q0 * Dn + (m0 + mt + 16 + 8 * half);
    *(float4*)(yp)     = make_float4(acc10[0], acc10[1], acc10[2], acc10[3]);
    *(float4*)(yp + 4) = make_float4(acc10[4], acc10[5], acc10[6], acc10[7]);

    yp = ybase + (size_t)q1 * Dn + (m0 + mt + 16 + 8 * half);
    *(float4*)(yp)     = make_float4(acc11[0], acc11[1], acc11[2], acc11[3]);
    *(float4*)(yp + 4) = make_float4(acc11[4], acc11[5], acc11[6], acc11[7]);
}

// ---------------------------------------------------------------------------
extern "C" void kernel_launch(void* const* d_in, const int* in_sizes, int n_in,
                              void* d_out, int out_size, void* d_ws, size_t ws_size,
                              hipStream_t stream)
{
    (void)in_sizes; (void)n_in; (void)out_size; (void)ws_size;

    const float* x     = (const float*)d_in[0];
    const float* bn_g  = (const float*)d_in[1];
    const float* bn_b  = (const float*)d_in[2];
    const float* bn_rm = (const float*)d_in[3];
    const float* bn_rv = (const float*)d_in[4];

    float* ori  = (float*)d_ws;                      // [B, D, S]  128 MB
    float* outb = ori + (size_t)Bn * Dn * Sn;        // [B, D, S]  128 MB (reused)

    float* out = (float*)d_out;

    // output[0] = x  (tuple element 0 of the reference)
    hipMemcpyAsync(out, x, (size_t)Bn * Sn * Dn * sizeof(float),
                   hipMemcpyDeviceToDevice, stream);

    bn_transpose_kernel<<<dim3(Sn / 32, Dn / 32, Bn), dim3(32, 8), 0, stream>>>(
        x, bn_g, bn_b, bn_rm, bn_rv, ori);

    size_t yoff = (size_t)Bn * Sn * Dn;
    for (int j = 0; j < 3; ++j) {
        const float* g1  = (const float*)d_in[5 + 8 * j + 0];
        const float* b1  = (const float*)d_in[5 + 8 * j + 1];
        const float* rm1 = (const float*)d_in[5 + 8 * j + 2];
        const float* rv1 = (const float*)d_in[5 + 8 * j + 3];
        const float* Wa  = (const float*)d_in[5 + 8 * j + 4];
        const float* ba  = (const float*)d_in[5 + 8 * j + 5];
        const float* Wl  = (const float*)d_in[5 + 8 * j + 6];
        const float* blv = (const float*)d_in[5 + 8 * j + 7];

        const int P = 2 << j;          // 2, 4, 8
        const int Q = Sn / P;          // 256, 128, 64

        const int nblocks = (Bn * Dn) / 256;
        if (P == 2)
            scan_kernel<2><<<nblocks, 256, 0, stream>>>(ori, g1, b1, rm1, rv1, Wa, ba, outb);
        else if (P == 4)
            scan_kernel<4><<<nblocks, 256, 0, stream>>>(ori, g1, b1, rm1, rv1, Wa, ba, outb);
        else
            scan_kernel<8><<<nblocks, 256, 0, stream>>>(ori, g1, b1, rm1, rv1, Wa, ba, outb);

        wmma_gemm_kernel<<<dim3(Q / 64, Dn / 128, Bn), 256, 0, stream>>>(
            outb, Wl, blv, out + yoff, Q);

        yoff += (size_t)Bn * Q * Dn;
    }
}